// UnsortedSegmentExample_fixed_31439160606804
// MI455X (gfx1250) — compile-verified
//
#include <hip/hip_runtime.h>

typedef float v2f __attribute__((ext_vector_type(2)));
typedef float v4f __attribute__((ext_vector_type(4)));
typedef float v8f __attribute__((ext_vector_type(8)));

// ---------------------------------------------------------------------------
// Phase 1: zero the accumulators (sums in d_out, counts in d_ws).
// Must run every launch (harness poisons buffers once and never re-zeros).
// ---------------------------------------------------------------------------
__global__ void seg_zero_kernel(float* __restrict__ sums, float* __restrict__ counts,
                                int nSum, int nCnt) {
  int i = blockIdx.x * blockDim.x + threadIdx.x;
  int stride = gridDim.x * blockDim.x;
  int total = nSum + nCnt;
  for (; i < total; i += stride) {
    if (i < nSum) sums[i] = 0.0f;
    else          counts[i - nSum] = 0.0f;
  }
}

// ---------------------------------------------------------------------------
// Phase 2: bandwidth phase. Stream data with non-temporal 128-bit loads
// (keeps the 2 MB accumulator table hot in the 192 MB L2 while 264 MB
// streams past), scatter with global_atomic_add_f32 (no-return atomics).
// One thread per float4 (4 consecutive columns of one row).
// ---------------------------------------------------------------------------
__global__ void seg_accum_kernel(const v4f* __restrict__ data4,
                                 const int* __restrict__ segids,
                                 float* __restrict__ sums,
                                 float* __restrict__ counts,
                                 long long totalVec4, int vecPerRow, int D) {
  long long i = (long long)blockIdx.x * blockDim.x + threadIdx.x;
  if (i >= totalVec4) return;
  const int row = (int)(i / vecPerRow);
  const int cg  = (int)(i % vecPerRow);
  const int s   = segids[row];

  // Non-temporal: this stream is read exactly once; don't pollute caches.
  v4f v = __builtin_nontemporal_load(&data4[i]);

  float* dst = sums + (size_t)s * D + cg * 4;
  atomicAdd(dst + 0, v.x);
  atomicAdd(dst + 1, v.y);
  atomicAdd(dst + 2, v.z);
  atomicAdd(dst + 3, v.w);
  if (cg == 0) atomicAdd(counts + s, 1.0f);
}

// ---------------------------------------------------------------------------
// Phase 3: mean = diag(1/clamp(count,1)) @ sums, computed per 16x16 tile with
// V_WMMA_F32_16X16X4_F32 (full f32 precision, RNE). Four accumulating K=4
// passes cover the 16-row diagonal: pass p uses A[m,k] = (m==4p+k)?rcp[m]:0,
// B[k,n] = sums[segBase+4p+k, colBase+n].
//
// VGPR layouts per CDNA5 ISA 7.12.2:
//   A 16x4 f32 (2 VGPRs): lane L holds M=L%16; VGPR{0,1} hold K={0,1} for
//     lanes 0-15 and K={2,3} for lanes 16-31.
//   B 4x16 f32 (2 VGPRs): lane L holds N=L%16, same K split across halves.
//   C/D 16x16 f32 (8 VGPRs): VGPR j holds M=j (lanes 0-15) / M=j+8 (16-31).
// EXEC must be all 1s: grid is exact and the tile guard is wave-uniform.
// ---------------------------------------------------------------------------
__global__ void seg_scale_wmma_kernel(float* __restrict__ sums,
                                      const float* __restrict__ counts,
                                      int S, int D, int nTiles) {
  const int wave = (blockIdx.x * blockDim.x + threadIdx.x) >> 5;
  if (wave >= nTiles) return;                       // wave-uniform
  const int lane    = threadIdx.x & 31;
  const int tilesPerSegBlk = D >> 4;                // 16-col tiles per 16-seg block
  const int segBase = (wave / tilesPerSegBlk) << 4;
  const int colBase = (wave % tilesPerSegBlk) << 4;
  const int half    = lane >> 4;                    // 0: lanes 0-15, 1: lanes 16-31
  const int l16     = lane & 15;
  const int k0      = half * 2;                     // this half's K pair {k0,k0+1}

  // Per-lane reciprocal of clamped count for row M=l16 of this tile.
  float c = counts[segBase + l16];
  c = (c < 1.0f) ? 1.0f : c;
  const float r = 1.0f / c;

  v8f acc = {};
#pragma unroll
  for (int p = 0; p < 4; ++p) {
    v2f a, b;
    // Shifted diagonal block: A[m,k] = rcp[m] iff m == 4p+k.
    a.x = (l16 == 4 * p + k0    ) ? r : 0.0f;
    a.y = (l16 == 4 * p + k0 + 1) ? r : 0.0f;
    // B rows 4p+k0, 4p+k0+1 of the sum tile; lane supplies column N=l16.
    const float* srow = sums + (size_t)(segBase + 4 * p + k0) * D + colBase + l16;
    b.x = srow[0];
    b.y = srow[D];
    acc = __builtin_amdgcn_wmma_f32_16x16x4_f32(
        /*neg_a=*/false, a, /*neg_b=*/false, b,
        /*c_mod=*/(short)0, acc, /*reuse_a=*/false, /*reuse_b=*/false);
  }

  // Write the scaled tile back in place (each wave owns its tile exclusively).
#pragma unroll
  for (int j = 0; j < 8; ++j) {
    const int m = j + half * 8;
    sums[(size_t)(segBase + m) * D + colBase + l16] = acc[j];
  }
}

// ---------------------------------------------------------------------------
extern "C" void kernel_launch(void* const* d_in, const int* in_sizes, int n_in,
                              void* d_out, int out_size, void* d_ws, size_t ws_size,
                              hipStream_t stream) {
  const float* data   = (const float*)d_in[0];
  const int*   segids = (const int*)d_in[1];   // integer input -> const int*
  const int N = in_sizes[1];                   // 1048576
  const int D = in_sizes[0] / N;               // 64
  const int S = out_size / D;                  // 8192

  float* sums   = (float*)d_out;               // accumulate means in place
  float* counts = (float*)d_ws;                // S floats of scratch

  // Phase 1: zero sums + counts.
  {
    const int total   = out_size + S;
    const int threads = 256;
    const int blocks  = (total + threads - 1) / threads;
    seg_zero_kernel<<<blocks, threads, 0, stream>>>(sums, counts, out_size, S);
  }

  // Phase 2: streaming scatter-add (bandwidth bound; ~11.4 us roofline).
  {
    const int vecPerRow = D / 4;
    const long long totalVec4 = (long long)N * vecPerRow;
    const int threads = 256;
    const long long blocks = (totalVec4 + threads - 1) / threads;
    seg_accum_kernel<<<(unsigned)blocks, threads, 0, stream>>>(
        (const v4f*)data, segids, sums, counts, totalVec4, vecPerRow, D);
  }

  // Phase 3: WMMA row-scale (touches only 2 MB; cost is noise).
  {
    const int nTiles  = (S / 16) * (D / 16);   // 2048 tiles, one wave each
    const int threads = 256;                   // 8 waves per block
    const int blocks  = (nTiles * 32 + threads - 1) / threads;
    seg_scale_wmma_kernel<<<blocks, threads, 0, stream>>>(sums, counts, S, D, nTiles);
  }
}